// CustomGraphNetwork_82265803588044
// MI455X (gfx1250) — compile-verified
//
#include <hip/hip_runtime.h>

// ---------------------------------------------------------------------------
// GNN (2-layer message passing) for MI455X / gfx1250, FP32 WMMA path.
//   concat(x[src], ea) @ Wmsg  ==  x[src] @ Wmsg[:64]  +  ea @ Wmsg[64:]
// => precompute P = x @ Wmsg[:64] as a dense GEMM (WMMA, 5 M-tiles/wave with
//    B-fragment reuse), keep only the tiny 16xC edge matmul per edge (WMMA,
//    16 edges per wave, indices on the scalar-memory path), scatter with
//    coalesced f32 atomics straight into the self-term buffer.
// ---------------------------------------------------------------------------

typedef __attribute__((ext_vector_type(2))) float v2f;
typedef __attribute__((ext_vector_type(8))) float v8f;

#define NODES 50000
#define EDGES 800000

__device__ __forceinline__ v8f wmma_f32_16x16x4(v2f a, v2f b, v8f c) {
  // 8 args: (neg_a, A, neg_b, B, c_mod, C, reuse_a, reuse_b)
  return __builtin_amdgcn_wmma_f32_16x16x4_f32(false, a, false, b, (short)0, c,
                                               false, false);
}

// D_A = X(80x64) @ Wa(64xCA)          -> outA   (message projection P)
// D_B = X(80x64) @ Wb(64xCB) + biasB  -> outB   (self term, scatter target)
// 5 M-tiles (80 rows) per block; one 16-col N-tile per wave; each B fragment
// is loaded once per K-step and reused by 5 independent accumulators.
template <int CA, int CB>
__global__ __launch_bounds__(32 * (CA / 16 + CB / 16))
void node_gemm_dual(const float* __restrict__ X,
                    const float* __restrict__ Wa,
                    const float* __restrict__ Wb,
                    const float* __restrict__ biasB,
                    float* __restrict__ outA,
                    float* __restrict__ outB) {
  const int wave  = threadIdx.x >> 5;
  const int lane  = threadIdx.x & 31;
  const int mrow  = lane & 15;   // A-matrix row held by this lane
  const int khalf = lane >> 4;   // which K-pair (0 -> K0/K1, 1 -> K2/K3)
  const int nn    = lane & 15;   // B/C/D column held by this lane
  const int m0    = blockIdx.x * 80;

  constexpr int NT_A = CA / 16;
  const bool   partB = (wave >= NT_A);
  const float* W     = partB ? Wb : Wa;
  float*       out   = partB ? outB : outA;
  const int    C     = partB ? CB : CA;
  const int    n0    = (partB ? (wave - NT_A) : wave) << 4;

  v8f acc[5];
  const float binit = partB ? biasB[n0 + nn] : 0.0f;
#pragma unroll
  for (int mt = 0; mt < 5; ++mt)
#pragma unroll
    for (int r = 0; r < 8; ++r) acc[mt][r] = binit;

  const float* xbase = X + (size_t)(m0 + mrow) * 64 + 2 * khalf;
#pragma unroll
  for (int kc = 0; kc < 16; ++kc) {  // K = 64, 4 per WMMA
    const int ka = kc * 4 + 2 * khalf;
    v2f b;
    b.x = W[(size_t)ka * C + n0 + nn];
    b.y = W[(size_t)(ka + 1) * C + n0 + nn];
#pragma unroll
    for (int mt = 0; mt < 5; ++mt) {  // reuse B across 5 M-tiles
      const float* xp = xbase + (size_t)mt * 16 * 64 + kc * 4;
      v2f a;
      a.x = xp[0];
      a.y = xp[1];
      acc[mt] = wmma_f32_16x16x4(a, b, acc[mt]);
    }
  }
#pragma unroll
  for (int mt = 0; mt < 5; ++mt)
#pragma unroll
    for (int r = 0; r < 8; ++r)
      out[(size_t)(m0 + mt * 16 + r + 8 * khalf) * C + n0 + nn] = acc[mt][r];
}

// Per-edge: q = ea(16x16) @ We(16xC) + bias (WMMA, 16 edges per wave),
// then AGG[dst] += q + P[src].  Edge indices are wave-uniform -> forced onto
// the scalar (SMEM/KMcnt) path via readfirstlane so the vector pipe only
// carries the divergent gathers and atomics.
template <int C>
__global__ __launch_bounds__(256)
void edge_msg_scatter(const long long* __restrict__ src,
                      const long long* __restrict__ dst,
                      const float* __restrict__ ea,    // E x 16
                      const float* __restrict__ We,    // 16 x C
                      const float* __restrict__ bias,  // C
                      const float* __restrict__ P,     // NODES x C
                      float* __restrict__ AGG) {       // NODES x C
  const int wave  = threadIdx.x >> 5;
  const int lane  = threadIdx.x & 31;
  const int mrow  = lane & 15;
  const int khalf = lane >> 4;
  const int nn    = lane & 15;
  // wave-uniform base edge; readfirstlane pins it in an SGPR
  const int e0 = __builtin_amdgcn_readfirstlane((blockIdx.x * 8 + wave) << 4);

  // Edge indices: uniform addresses -> scalar loads.
  long long us[16], ud[16];
#pragma unroll
  for (int m = 0; m < 16; ++m) {
    us[m] = src[e0 + m];
    ud[m] = dst[e0 + m];
  }

  // A fragments: K = 16 (4 WMMA K-steps), hoisted once per wave.
  v2f afrag[4];
  const float* ap = ea + (size_t)(e0 + mrow) * 16 + 2 * khalf;
#pragma unroll
  for (int kc = 0; kc < 4; ++kc) {
    afrag[kc].x = ap[kc * 4 + 0];
    afrag[kc].y = ap[kc * 4 + 1];
  }

#pragma unroll
  for (int nt = 0; nt < C / 16; ++nt) {
    const int n0 = nt << 4;
    v8f       acc;
    const float binit = bias[n0 + nn];
#pragma unroll
    for (int r = 0; r < 8; ++r) acc[r] = binit;
#pragma unroll
    for (int kc = 0; kc < 4; ++kc) {
      const int ka = kc * 4 + 2 * khalf;
      v2f       b;
      b.x = We[(size_t)ka * C + n0 + nn];
      b.y = We[(size_t)(ka + 1) * C + n0 + nn];
      acc = wmma_f32_16x16x4(afrag[kc], b, acc);
    }
#pragma unroll
    for (int r = 0; r < 8; ++r) {
      // lane's acc[r] belongs to edge row m = r + 8*khalf: pick between the
      // two uniform index values with a cndmask.
      const long long si = khalf ? us[r + 8] : us[r];
      const long long di = khalf ? ud[r + 8] : ud[r];
      const float v = acc[r] + P[(size_t)si * C + n0 + nn];
      atomicAdd(&AGG[(size_t)di * C + n0 + nn], v);
    }
  }
}

__global__ __launch_bounds__(256)
void bn_relu_inplace(float* __restrict__ h,
                     const float* __restrict__ gamma,
                     const float* __restrict__ beta,
                     const float* __restrict__ mean,
                     const float* __restrict__ var, int total4) {
  const int i = blockIdx.x * 256 + threadIdx.x;
  if (i < total4) {
    const int c0 = (i * 4) & 63;  // 4 consecutive channels within a 64-row
    float4 v = ((float4*)h)[i];
    float* vp = &v.x;
#pragma unroll
    for (int j = 0; j < 4; ++j) {
      float t = (vp[j] - mean[c0 + j]) * rsqrtf(var[c0 + j] + 1e-5f) *
                    gamma[c0 + j] +
                beta[c0 + j];
      vp[j] = t > 0.0f ? t : 0.0f;
    }
    ((float4*)h)[i] = v;
  }
}

extern "C" void kernel_launch(void* const* d_in, const int* in_sizes, int n_in,
                              void* d_out, int out_size, void* d_ws,
                              size_t ws_size, hipStream_t stream) {
  const float*     x       = (const float*)d_in[0];
  const long long* ei      = (const long long*)d_in[1];  // int64 [2,E]
  const float*     ea      = (const float*)d_in[2];
  const float*     W1_msg  = (const float*)d_in[3];   // 80x64 row-major
  const float*     b1_msg  = (const float*)d_in[4];
  const float*     W1_self = (const float*)d_in[5];   // 64x64
  const float*     b1_self = (const float*)d_in[6];
  const float*     bn_g    = (const float*)d_in[7];
  const float*     bn_b    = (const float*)d_in[8];
  const float*     bn_m    = (const float*)d_in[9];
  const float*     bn_v    = (const float*)d_in[10];
  const float*     W2_msg  = (const float*)d_in[11];  // 80x32
  const float*     b2_msg  = (const float*)d_in[12];
  const float*     W2_self = (const float*)d_in[13];  // 64x32
  const float*     b2_self = (const float*)d_in[14];

  const long long* src = ei;
  const long long* dst = ei + EDGES;

  float* P1  = (float*)d_ws;                  // NODES*64  (x @ W1_msg[:64])
  float* S1  = P1 + (size_t)NODES * 64;       // NODES*64  (self + scatter + BN)
  float* P2  = S1 + (size_t)NODES * 64;       // NODES*32  (h @ W2_msg[:64])
  float* out = (float*)d_out;                 // NODES*32

  // Layer 1: node GEMMs (P1, and S1 = x@W1_self + b1_self)
  node_gemm_dual<64, 64><<<NODES / 80, 256, 0, stream>>>(
      x, W1_msg, W1_self, b1_self, P1, S1);
  // Layer 1 edges: S1[dst] += P1[src] + ea @ W1_msg[64:] + b1_msg
  edge_msg_scatter<64><<<EDGES / 128, 256, 0, stream>>>(
      src, dst, ea, W1_msg + 64 * 64, b1_msg, P1, S1);
  // BatchNorm (running stats) + ReLU, in place: S1 becomes h
  bn_relu_inplace<<<(NODES * 16 + 255) / 256, 256, 0, stream>>>(
      S1, bn_g, bn_b, bn_m, bn_v, NODES * 16);
  // Layer 2: node GEMMs (P2, and d_out = h@W2_self + b2_self)
  node_gemm_dual<32, 32><<<NODES / 80, 128, 0, stream>>>(
      S1, W2_msg, W2_self, b2_self, P2, out);
  // Layer 2 edges: d_out[dst] += P2[src] + ea @ W2_msg[64:] + b2_msg
  edge_msg_scatter<32><<<EDGES / 128, 256, 0, stream>>>(
      src, dst, ea, W2_msg + 64 * 32, b2_msg, P2, out);
}